// TPSGridGenerator_35871566856426
// MI455X (gfx1250) — compile-verified
//
#include <hip/hip_runtime.h>
#include <math.h>

// ---------------------------------------------------------------------------
// TPS grid generator for MI455X (gfx1250, wave32, WMMA).
//
//   out[b, h, w, {x,y}] = A0 + A1*gx + A2*gy + sum_n U(p,n) * W[b,n]
//
// Cast as GEMM with 28-wide basis Uext[p, k] (25 TPS kernels + {1,gx,gy})
// against per-batch coefficients Wext[ch][b][k], evaluated with
// V_WMMA_F32_16X16X4_F32 (K=28 = 7 steps of 4).
//
// Roofline: ~1.4 GFLOP of contraction vs ~100 MB of streamed output
// -> store-bound (~4.3 us at 23.3 TB/s). Hence: WMMA for the math,
// coalesced b64 {X,Y} pair stores with TH=NT (output is never re-read;
// keep L2 clean for the hot 57 KB coefficient table).
// ---------------------------------------------------------------------------

typedef __attribute__((ext_vector_type(2))) float v2f;
typedef __attribute__((ext_vector_type(8))) float v8f;

#define OUT_H 256
#define OUT_W 192
#define HW (OUT_H * OUT_W)          // 49152 pixels = 3072 tiles of 16
#define NCTRL 25
#define KDIM 28                      // 25 basis + affine {1, gx, gy}
#define BATCH 256

__device__ __forceinline__ float ctrl_x(int n) { return -1.0f + 0.5f * (float)(n / 5); }
__device__ __forceinline__ float ctrl_y(int n) { return -1.0f + 0.5f * (float)(n % 5); }

// TPS radial basis U(r) = d2 * ln(d2), with d2==0 -> 1 (so U=0 per reference).
__device__ __forceinline__ float tps_u(float gx, float gy, int n) {
    float dx = gx - ctrl_x(n);
    float dy = gy - ctrl_y(n);
    float d2 = dx * dx + dy * dy;
    d2 = (d2 == 0.0f) ? 1.0f : d2;
    return d2 * logf(d2);
}

__device__ __forceinline__ float basis(float gx, float gy, int k) {
    if (k < NCTRL) return tps_u(gx, gy, k);
    if (k == 25) return 1.0f;
    if (k == 26) return gx;
    return gy;
}

// ---------------------------------------------------------------------------
// Stage 0: build the 28x28 TPS system matrix L and invert it (Gauss-Jordan
// with partial pivoting; the bottom-right 3x3 block of L is zero so pivoting
// is mandatory). One wave32 workgroup; trivial cost.
// ---------------------------------------------------------------------------
__global__ __launch_bounds__(32) void tps_invert(float* __restrict__ li) {
    __shared__ float M[KDIM][57];   // augmented [L | I], padded row
    __shared__ int piv;
    const int lane = threadIdx.x;

    for (int c = lane; c < 2 * KDIM; c += 32) {
        for (int r = 0; r < KDIM; ++r) {
            float v;
            if (c >= KDIM) {
                v = ((c - KDIM) == r) ? 1.0f : 0.0f;
            } else if (r < NCTRL && c < NCTRL) {
                float dx = ctrl_x(r) - ctrl_x(c);
                float dy = ctrl_y(r) - ctrl_y(c);
                float d2 = dx * dx + dy * dy;
                d2 = (d2 == 0.0f) ? 1.0f : d2;
                v = d2 * logf(d2);
            } else if (r < NCTRL) {            // P block: [1, Px, Py]
                v = (c == 25) ? 1.0f : (c == 26) ? ctrl_x(r) : ctrl_y(r);
            } else if (c < NCTRL) {            // P^T block
                v = (r == 25) ? 1.0f : (r == 26) ? ctrl_x(c) : ctrl_y(c);
            } else {
                v = 0.0f;                      // zero corner
            }
            M[r][c] = v;
        }
    }
    __syncthreads();

    for (int k = 0; k < KDIM; ++k) {
        if (lane == 0) {
            int best = k;
            float bm = fabsf(M[k][k]);
            for (int r = k + 1; r < KDIM; ++r) {
                float a = fabsf(M[r][k]);
                if (a > bm) { bm = a; best = r; }
            }
            piv = best;
        }
        __syncthreads();
        const int pr = piv;
        for (int c = lane; c < 2 * KDIM; c += 32) {
            float t = M[k][c]; M[k][c] = M[pr][c]; M[pr][c] = t;
        }
        __syncthreads();
        const float inv = 1.0f / M[k][k];
        for (int c = lane; c < 2 * KDIM; c += 32) M[k][c] *= inv;
        __syncthreads();
        // Row-parallel elimination: lane r owns row r (no LDS races).
        for (int r = lane; r < KDIM; r += 32) {
            if (r != k) {
                float f = M[r][k];
                for (int c = 0; c < 2 * KDIM; ++c) M[r][c] -= f * M[k][c];
            }
        }
        __syncthreads();
    }

    for (int c = lane; c < KDIM; c += 32)
        for (int r = 0; r < KDIM; ++r)
            li[r * KDIM + c] = M[r][c + KDIM];
}

// ---------------------------------------------------------------------------
// Stage 1: Wext[ch][b][k] = sum_{m<25} Li[k][m] * (theta[b][ch*25+m] + P[m])
// Rows 0..24 are the warp weights W, rows 25..27 are the affine A coeffs.
// 2*28*256 = 14336 independent dot products of length 25.
// ---------------------------------------------------------------------------
__global__ __launch_bounds__(256) void tps_coeff(const float* __restrict__ theta,
                                                 const float* __restrict__ li,
                                                 float* __restrict__ wext) {
    const int idx = blockIdx.x * 256 + threadIdx.x;       // < 2*28*256
    const int ch  = idx / (KDIM * BATCH);
    const int rem = idx % (KDIM * BATCH);
    const int k   = rem / BATCH;
    const int b   = rem % BATCH;

    const float* th = theta + (size_t)b * (2 * NCTRL) + ch * NCTRL;
    float acc = 0.0f;
#pragma unroll
    for (int m = 0; m < NCTRL; ++m) {
        float pm = (ch == 0) ? ctrl_x(m) : ctrl_y(m);
        acc += li[k * KDIM + m] * (th[m] + pm);
    }
    wext[((size_t)ch * BATCH + b) * KDIM + k] = acc;
}

// ---------------------------------------------------------------------------
// Stage 2: the GEMM. out[b,p] = sum_k Wext[b,k] * Uext[p,k] via
// V_WMMA_F32_16X16X4_F32. One wave per 16-pixel tile; the 7 B-operand
// register pairs (basis) are computed once and reused across all 16 batch
// tiles x 2 channels (224 WMMAs per wave). Stores are coalesced b64 {X,Y}
// pairs, non-temporal (output is write-once) -> HBM-write bound.
//
// Operand lane layouts (ISA 7.12.2, 32-bit):
//   A 16x4 : lanes 0-15 rows M, v0=K0 v1=K1 ; lanes 16-31 v0=K2 v1=K3
//   B 4x16 : lanes hold N=lane&15 ; lo half K={4t,4t+1}, hi half K={4t+2,4t+3}
//   C 16x16: VGPR r -> row r (lanes 0-15) / row r+8 (lanes 16-31), N=lane&15
// ---------------------------------------------------------------------------
__global__ __launch_bounds__(256) void tps_main(const float* __restrict__ wext,
                                                float* __restrict__ out) {
    const int lane = threadIdx.x & 31;
    const int wave = threadIdx.x >> 5;
    const int tile = blockIdx.x * 8 + wave;   // 0..3071, exact
    const int lp   = lane & 15;
    const int hi   = lane >> 4;               // 0 or 1

    const int p = tile * 16 + lp;             // this lane's pixel (N index)
    const int w = p % OUT_W;
    const int h = p / OUT_W;
    const float gx = -1.0f + (2.0f / (OUT_W - 1)) * (float)w;
    const float gy = -1.0f + (2.0f / (OUT_H - 1)) * (float)h;

    // B operand: basis values in WMMA B lane layout, reused for every tile.
    v2f bk[7];
#pragma unroll
    for (int t = 0; t < 7; ++t) {
        const int k0 = 4 * t + 2 * hi;
        bk[t].x = basis(gx, gy, k0);
        bk[t].y = basis(gx, gy, k0 + 1);
    }

    const v2f* wx = (const v2f*)(wext);                         // [256][14] pairs
    const v2f* wy = (const v2f*)(wext + (size_t)BATCH * KDIM);

    for (int b0 = 0; b0 < BATCH; b0 += 16) {
        const int row = b0 + lp;              // A-operand row (M index)
        const int cb  = hi;                   // pair offset: K+2 for hi half

        v8f cx = {};
        v8f cy = {};
#pragma unroll
        for (int t = 0; t < 7; ++t) {
            v2f a = wx[row * 7 * 2 + 2 * t + cb];
            cx = __builtin_amdgcn_wmma_f32_16x16x4_f32(
                false, a, false, bk[t], (short)0, cx, false, false);
        }
#pragma unroll
        for (int t = 0; t < 7; ++t) {
            v2f a = wy[row * 7 * 2 + 2 * t + cb];
            cy = __builtin_amdgcn_wmma_f32_16x16x4_f32(
                false, a, false, bk[t], (short)0, cy, false, false);
        }

        // Store interleaved {X, Y} pairs: out[((b*HW + p)*2 + c)].
        // Non-temporal b64 stores: the 100 MB output is never re-read on
        // device; don't wash the coefficient table out of L2.
#pragma unroll
        for (int r = 0; r < 8; ++r) {
            const int b = b0 + r + hi * 8;
            v2f v;
            v.x = cx[r];
            v.y = cy[r];
            __builtin_nontemporal_store(v, (v2f*)(out + ((size_t)b * HW + (size_t)p) * 2));
        }
    }
}

// ---------------------------------------------------------------------------
extern "C" void kernel_launch(void* const* d_in, const int* in_sizes, int n_in,
                              void* d_out, int out_size, void* d_ws, size_t ws_size,
                              hipStream_t stream) {
    const float* theta = (const float*)d_in[0];   // (256, 50) fp32
    float* out = (float*)d_out;                   // (256, 256, 192, 2) fp32

    // Workspace layout (fp32): [ Li 28x28 | Wext 2x256x28 ]  (~60 KB)
    float* li   = (float*)d_ws;
    float* wext = li + KDIM * KDIM;

    tps_invert<<<1, 32, 0, stream>>>(li);
    tps_coeff<<<(2 * KDIM * BATCH) / 256, 256, 0, stream>>>(theta, li, wext);
    tps_main<<<(HW / 16) / 8, 256, 0, stream>>>(wext, out);
}